// BusStopPredictor_62165356642602
// MI455X (gfx1250) — compile-verified
//
#include <hip/hip_runtime.h>
#include <hip/hip_bf16.h>
#include <math.h>

typedef __attribute__((ext_vector_type(2))) float v2f;
typedef __attribute__((ext_vector_type(8))) float v8f;

// ---------------------------------------------------------------------------
// Degree accumulation: deg[dst] += 1 per edge (self-loop +1 added later).
// ---------------------------------------------------------------------------
__global__ void deg_kernel(const int* __restrict__ eidx, float* __restrict__ deg, int E) {
    int e = blockIdx.x * blockDim.x + threadIdx.x;
    if (e >= E) return;
    int d = eidx[E + e];           // dst row of edge_index [2,E]
    atomicAdd(&deg[d], 1.0f);
}

// deg -> dinv = rsqrt(deg + 1)  (self loop guarantees deg+1 >= 1)
__global__ void dinv_kernel(float* __restrict__ dinv, int N) {
    int i = blockIdx.x * blockDim.x + threadIdx.x;
    if (i >= N) return;
    dinv[i] = rsqrtf(dinv[i] + 1.0f);
}

// ---------------------------------------------------------------------------
// h1 = x @ W1   (K=2; one thread per (node, 4-col chunk))
// ---------------------------------------------------------------------------
__global__ void h1_kernel(const float* __restrict__ x, const float* __restrict__ W1,
                          float* __restrict__ h1, int N) {
    int tid = blockIdx.x * blockDim.x + threadIdx.x;
    if (tid >= N * 32) return;
    int node = tid >> 5;
    int j = (tid & 31) * 4;
    float x0 = x[(size_t)node * 2 + 0];
    float x1 = x[(size_t)node * 2 + 1];
    float4 w0 = *(const float4*)&W1[j];
    float4 w1 = *(const float4*)&W1[128 + j];
    float4 o;
    o.x = x0 * w0.x + x1 * w1.x;
    o.y = x0 * w0.y + x1 * w1.y;
    o.z = x0 * w0.z + x1 * w1.z;
    o.w = x0 * w0.w + x1 * w1.w;
    *(float4*)&h1[(size_t)node * 128 + j] = o;
}

// ---------------------------------------------------------------------------
// Edge scatter-add, feature width F (multiple of 4): agg[dst] += h[src]*norm
// one thread per (edge, 4-feature chunk). Gathers hit L2 (h fits in 192MB L2).
// ---------------------------------------------------------------------------
template <int F>
__global__ void edge_kernel(const int* __restrict__ eidx, const float* __restrict__ dinv,
                            const float* __restrict__ h, float* __restrict__ agg, int E) {
    const int CH = F / 4;                       // chunks per edge
    int tid = blockIdx.x * blockDim.x + threadIdx.x;
    if (tid >= E * CH) return;
    int e = tid / CH;
    int j = (tid % CH) * 4;
    int s = eidx[e];
    int d = eidx[E + e];
    float norm = dinv[s] * dinv[d];
    float4 v = *(const float4*)&h[(size_t)s * F + j];
    float* o = &agg[(size_t)d * F + j];
    atomicAdd(o + 0, v.x * norm);
    atomicAdd(o + 1, v.y * norm);
    atomicAdd(o + 2, v.z * norm);
    atomicAdd(o + 3, v.w * norm);
}

// ---------------------------------------------------------------------------
// Post pass: out = relu(agg + h*dinv^2 + b), written into agg in place.
// ---------------------------------------------------------------------------
template <int F>
__global__ void post_kernel(const float* __restrict__ h, const float* __restrict__ dinv,
                            const float* __restrict__ b, float* __restrict__ agg, int N) {
    const int CH = F / 4;
    int tid = blockIdx.x * blockDim.x + threadIdx.x;
    if (tid >= N * CH) return;
    int node = tid / CH;
    int j = (tid % CH) * 4;
    float di = dinv[node];
    float sl = di * di;
    float4 a = *(const float4*)&agg[(size_t)node * F + j];
    float4 hv = *(const float4*)&h[(size_t)node * F + j];
    float4 bv = *(const float4*)&b[j];
    float4 o;
    o.x = fmaxf(a.x + hv.x * sl + bv.x, 0.0f);
    o.y = fmaxf(a.y + hv.y * sl + bv.y, 0.0f);
    o.z = fmaxf(a.z + hv.z * sl + bv.z, 0.0f);
    o.w = fmaxf(a.w + hv.w * sl + bv.w, 0.0f);
    *(float4*)&agg[(size_t)node * F + j] = o;
}

// ---------------------------------------------------------------------------
// h2 = out1 @ W2  : [N,128] x [128,64], fp32 WMMA 16x16x4.
// Block = 256 threads = 8 waves. Each wave computes one 16x16 tile of C;
// block covers 32 rows x 64 cols. W2 (32KB) + A tile (16KB) staged in LDS.
//
// Per ISA layouts (CDNA5 ISA):
//   A 16x4 f32:  lane<16 holds {K0,K1} for M=lane; lane>=16 holds {K2,K3}.
//   B 4x16 f32:  mirrored — lane-half selects K pair, lane&15 = N.
//   C/D 16x16:   VGPR v -> M = v + 8*(lane>=16), N = lane&15.
// ---------------------------------------------------------------------------
__global__ __launch_bounds__(256) void gemm2_wmma(const float* __restrict__ A,  // out1 [N,128]
                                                  const float* __restrict__ W,  // W2 [128,64]
                                                  float* __restrict__ C,        // h2 [N,64]
                                                  int N) {
    __shared__ float sW[128 * 64];
    __shared__ float sA[32 * 128];
    int t = threadIdx.x;
    int row0 = blockIdx.x * 32;

    // stage W2: 8192 floats = 2048 float4
    const float4* W4 = (const float4*)W;
    float4* sW4 = (float4*)sW;
    for (int i = t; i < 2048; i += 256) sW4[i] = W4[i];

    // stage 32 rows of A: 4096 floats = 1024 float4 (32 float4 per row)
    float4* sA4 = (float4*)sA;
    for (int i = t; i < 1024; i += 256) {
        int r = row0 + (i >> 5);
        float4 v = make_float4(0.f, 0.f, 0.f, 0.f);
        if (r < N) v = ((const float4*)A)[(size_t)r * 32 + (i & 31)];
        sA4[i] = v;
    }
    __syncthreads();

    int lane  = t & 31;
    int warp  = t >> 5;
    int lrow  = lane & 15;
    int lhalf = lane >> 4;          // 0: K pair {0,1}, 1: K pair {2,3}
    int rloc  = (warp >> 2) * 16;   // 0 or 16 within the 32-row tile
    int col0  = (warp & 3) * 16;    // 0,16,32,48

    const float* aRow = &sA[(size_t)(rloc + lrow) * 128 + 2 * lhalf];
    const float* bCol = &sW[(size_t)(2 * lhalf) * 64 + col0 + lrow];

    v8f acc = {};
    #pragma unroll
    for (int k = 0; k < 128; k += 4) {
        v2f a; a[0] = aRow[k];      a[1] = aRow[k + 1];
        v2f b; b[0] = bCol[k * 64]; b[1] = bCol[k * 64 + 64];
        acc = __builtin_amdgcn_wmma_f32_16x16x4_f32(
            /*neg_a=*/false, a, /*neg_b=*/false, b,
            /*c_mod=*/(short)0, acc, /*reuse_a=*/false, /*reuse_b=*/false);
    }

    int gcol = col0 + lrow;
    #pragma unroll
    for (int v = 0; v < 8; ++v) {
        int r = row0 + rloc + v + 8 * lhalf;
        if (r < N) C[(size_t)r * 64 + gcol] = acc[v];
    }
}

// ---------------------------------------------------------------------------
// head: out[i] = sigmoid(dot(out2[i], Wp) + bp)
// ---------------------------------------------------------------------------
__global__ void head_kernel(const float* __restrict__ h, const float* __restrict__ Wp,
                            const float* __restrict__ bp, float* __restrict__ out, int N) {
    int i = blockIdx.x * blockDim.x + threadIdx.x;
    if (i >= N) return;
    const float4* row = (const float4*)&h[(size_t)i * 64];
    const float4* w4  = (const float4*)Wp;
    float acc = 0.0f;
    #pragma unroll
    for (int k = 0; k < 16; ++k) {
        float4 v = row[k];
        float4 w = w4[k];
        acc += v.x * w.x + v.y * w.y + v.z * w.z + v.w * w.w;
    }
    float z = acc + bp[0];
    out[i] = 1.0f / (1.0f + expf(-z));
}

extern "C" void kernel_launch(void* const* d_in, const int* in_sizes, int n_in,
                              void* d_out, int out_size, void* d_ws, size_t ws_size,
                              hipStream_t stream) {
    const float* x  = (const float*)d_in[0];
    const int*   ei = (const int*)d_in[1];   // [2,E] int32
    const float* W1 = (const float*)d_in[2];
    const float* b1 = (const float*)d_in[3];
    const float* W2 = (const float*)d_in[4];
    const float* b2 = (const float*)d_in[5];
    const float* Wp = (const float*)d_in[6];
    const float* bp = (const float*)d_in[7];
    float* out = (float*)d_out;

    const int N = in_sizes[0] / 2;           // x is [N,2]
    const int E = in_sizes[1] / 2;           // edge_index is [2,E]

    // Workspace layout (floats): dinv[N] | buf0[N*128] | buf1[N*128]
    float* ws   = (float*)d_ws;
    float* dinv = ws;
    float* buf0 = ws + (size_t)N;                    // h1, later h2 (first N*64) + agg2 (second N*64)
    float* buf1 = buf0 + (size_t)N * 128;            // agg1 -> out1
    float* h1   = buf0;
    float* agg1 = buf1;
    float* h2   = buf0;                              // [N,64]
    float* agg2 = buf0 + (size_t)N * 64;             // [N,64]

    const int TB = 256;
    auto blocks = [](long long work, int tb) { return (unsigned)((work + tb - 1) / tb); };

    // 1) degrees -> dinv
    hipMemsetAsync(dinv, 0, (size_t)N * sizeof(float), stream);
    deg_kernel<<<blocks(E, TB), TB, 0, stream>>>(ei, dinv, E);
    dinv_kernel<<<blocks(N, TB), TB, 0, stream>>>(dinv, N);

    // 2) h1 = x @ W1
    h1_kernel<<<blocks((long long)N * 32, TB), TB, 0, stream>>>(x, W1, h1, N);

    // 3) layer-1 aggregation
    hipMemsetAsync(agg1, 0, (size_t)N * 128 * sizeof(float), stream);
    edge_kernel<128><<<blocks((long long)E * 32, TB), TB, 0, stream>>>(ei, dinv, h1, agg1, E);
    post_kernel<128><<<blocks((long long)N * 32, TB), TB, 0, stream>>>(h1, dinv, b1, agg1, N);
    // agg1 now holds out1; buf0 (h1) is free.

    // 4) h2 = out1 @ W2 via fp32 WMMA; zero agg2 region (disjoint half of buf0)
    hipMemsetAsync(agg2, 0, (size_t)N * 64 * sizeof(float), stream);
    gemm2_wmma<<<blocks(((long long)N + 31) / 32, 1), 256, 0, stream>>>(agg1, W2, h2, N);

    // 5) layer-2 aggregation
    edge_kernel<64><<<blocks((long long)E * 16, TB), TB, 0, stream>>>(ei, dinv, h2, agg2, E);
    post_kernel<64><<<blocks((long long)N * 16, TB), TB, 0, stream>>>(h2, dinv, b2, agg2, N);
    // agg2 now holds out2.

    // 6) prediction head
    head_kernel<<<blocks(N, TB), TB, 0, stream>>>(agg2, Wp, bp, out, N);
}